// SplitMemoryModel_3204045603520
// MI455X (gfx1250) — compile-verified
//
#include <hip/hip_runtime.h>
#include <hip/hip_bf16.h>

#define HID  64
#define VOC  64
#define HALF 32
#define BB   128
#define LL   2048

typedef float v2f __attribute__((ext_vector_type(2)));
typedef float v8f __attribute__((ext_vector_type(8)));

// Workspace layout (floats):
//  [    0, 2048)  k_s table  (64 vocab x 32)
//  [ 2048, 4096)  k_e table  (64 vocab x 32)
//  [ 4096, 4160)  rs = 1/(|k_s|^2 + 1e-6)   (64)
//  [ 4160, 4224)  re = 1/(|k_e|^2 + 1e-6)   (64)
//  [ 4224,12416)  ctx (128 x 64)  [sem | epi]
#define WS_KS   0
#define WS_KE   2048
#define WS_RS   4096
#define WS_RE   4160
#define WS_CTX  4224

// ---------------------------------------------------------------------------
// Kernel 1: per-vocab front end.  h = LN(e + MLP(e)); k tables; 1/den tables.
// One block per vocab id, 128 threads.
// ---------------------------------------------------------------------------
__global__ void precompute_tables(const float* __restrict__ embed,
                                  const float* __restrict__ W1, const float* __restrict__ b1,
                                  const float* __restrict__ W2, const float* __restrict__ b2,
                                  const float* __restrict__ ln_g, const float* __restrict__ ln_b,
                                  const float* __restrict__ Ws, const float* __restrict__ bs,
                                  const float* __restrict__ We, const float* __restrict__ be,
                                  float* __restrict__ ws)
{
    __shared__ float e[HID];
    __shared__ float hid[2 * HID];
    __shared__ float x[HID];
    __shared__ float h[HID];
    __shared__ float kk[2 * HALF];

    const int v   = blockIdx.x;
    const int tid = threadIdx.x;   // 0..127

    if (tid < HID) e[tid] = embed[v * HID + tid];
    __syncthreads();

    // hidden = relu(e @ W1 + b1), W1 is (64,128) row-major
    {
        float acc = b1[tid];
        #pragma unroll 8
        for (int i = 0; i < HID; ++i) acc = fmaf(e[i], W1[i * (2 * HID) + tid], acc);
        hid[tid] = acc > 0.f ? acc : 0.f;
    }
    __syncthreads();

    // x = e + hidden @ W2 + b2, W2 is (128,64)
    if (tid < HID) {
        float acc = b2[tid];
        #pragma unroll 8
        for (int i = 0; i < 2 * HID; ++i) acc = fmaf(hid[i], W2[i * HID + tid], acc);
        x[tid] = e[tid] + acc;
    }
    __syncthreads();

    // LayerNorm over 64 features (redundant per-thread reduction; trivial size)
    if (tid < HID) {
        float mu = 0.f;
        for (int i = 0; i < HID; ++i) mu += x[i];
        mu *= (1.0f / HID);
        float var = 0.f;
        for (int i = 0; i < HID; ++i) { float d = x[i] - mu; var = fmaf(d, d, var); }
        var *= (1.0f / HID);
        float inv = rsqrtf(var + 1e-5f);
        h[tid] = (x[tid] - mu) * inv * ln_g[tid] + ln_b[tid];
    }
    __syncthreads();

    // k_s = h @ Ws + bs ; k_e = h @ We + be   (Ws/We are (64,32))
    if (tid < HALF) {
        float acc = bs[tid];
        #pragma unroll 8
        for (int i = 0; i < HID; ++i) acc = fmaf(h[i], Ws[i * HALF + tid], acc);
        kk[tid] = acc;
        ws[WS_KS + v * HALF + tid] = acc;
    } else if (tid < 2 * HALF) {
        const int j = tid - HALF;
        float acc = be[j];
        #pragma unroll 8
        for (int i = 0; i < HID; ++i) acc = fmaf(h[i], We[i * HALF + j], acc);
        kk[tid] = acc;
        ws[WS_KE + v * HALF + j] = acc;
    }
    __syncthreads();

    if (tid == 0) {
        float ds = 0.f, de = 0.f;
        for (int i = 0; i < HALF; ++i) {
            ds = fmaf(kk[i], kk[i], ds);
            de = fmaf(kk[HALF + i], kk[HALF + i], de);
        }
        ws[WS_RS + v] = 1.0f / (ds + 1e-6f);
        ws[WS_RE + v] = 1.0f / (de + 1e-6f);
    }
}

// ---------------------------------------------------------------------------
// Kernel 2: adjoint scan.  M_new = M (I - w r k k^T) + w k k^T  implies
//   ctx = M_final q = sum_t w_t k_t (k_t . u_{t+1}),
//   u_t = u_{t+1} - w_t r_t (k_t . u_{t+1}) k_t,  u_T = q.
// One block per batch element; wave0 = semantic memory, wave1 = episodic.
// Lane i owns u[i], ctx[i]; dot via 5-level wave32 shuffle tree.
// ---------------------------------------------------------------------------
__global__ void scan_kernel(const int* __restrict__ seq,
                            float* __restrict__ ws)
{
    __shared__ float k_lds[2 * VOC * HALF];  // 16 KB: [0]=k_s table, [2048]=k_e
    __shared__ float r_lds[2 * VOC];         // rs | re
    __shared__ int   tok_lds[LL];            // 8 KB token row

    const int b   = blockIdx.x;
    const int tid = threadIdx.x;             // 0..63

    for (int i = tid; i < 2 * VOC * HALF; i += 64) k_lds[i] = ws[WS_KS + i];
    for (int i = tid; i < 2 * VOC;        i += 64) r_lds[i] = ws[WS_RS + i];
    const int* srow = seq + (size_t)b * LL;
    for (int i = tid; i < LL; i += 64) tok_lds[i] = srow[i];
    __syncthreads();

    const int half = tid >> 5;               // 0 = sem, 1 = epi
    const int lane = tid & 31;
    const float* kt = k_lds + half * (VOC * HALF);
    const float* rt = r_lds + half * VOC;

    const int tok_last = tok_lds[LL - 1];    // q row maps through same table
    float u   = kt[tok_last * HALF + lane];
    float ctx = 0.f;
    const float inv_L = 1.0f / (float)LL;

    for (int t = LL - 2; t >= 0; --t) {
        const int tok = tok_lds[t];                 // LDS broadcast
        const float k = kt[tok * HALF + lane];      // stride-1: conflict-free
        const float r = rt[tok];
        float d = k * u;                            // wave32 sum reduction
        d += __shfl_xor(d, 16, 32);
        d += __shfl_xor(d,  8, 32);
        d += __shfl_xor(d,  4, 32);
        d += __shfl_xor(d,  2, 32);
        d += __shfl_xor(d,  1, 32);
        const float w  = half ? (float)(t + 1) * inv_L : 1.0f;
        const float wd = w * d;
        ctx = fmaf(wd, k, ctx);
        u   = fmaf(-wd * r, k, u);
    }
    ws[WS_CTX + b * (2 * HALF) + half * HALF + lane] = ctx;
}

// ---------------------------------------------------------------------------
// Kernel 3: out = ctx @ Wo + bo via V_WMMA_F32_16X16X4_F32 (full f32).
// Grid (8,4): one wave per 16x16 output tile, 16 K-steps of K=4.
// A layout: lane row = lane&15, VGPR pair holds K = k0 + 2*(lane>=16) + {0,1}.
// B layout: lane col = lane&15, VGPR pair holds same K split.
// C/D: VGPR i -> M = (lane<16 ? i : 8+i), N = lane&15.
// ---------------------------------------------------------------------------
__global__ void out_gemm_wmma(const float* __restrict__ ws,
                              const float* __restrict__ Wo,   // (64,64)
                              const float* __restrict__ bo,   // (64)
                              float* __restrict__ out)        // (128,64)
{
    const float* ctx = ws + WS_CTX;
    const int lane  = threadIdx.x;            // 0..31
    const int mt    = blockIdx.x;             // 0..7  (M tiles of 128)
    const int nt    = blockIdx.y;             // 0..3  (N tiles of 64)
    const int row   = lane & 15;
    const int col   = lane & 15;
    const int khalf = (lane >> 4) << 1;       // 0 or 2
    const int m0 = mt * 16;
    const int n0 = nt * 16;

    v8f c;
    const float bias = bo[n0 + col];
    #pragma unroll
    for (int i = 0; i < 8; ++i) c[i] = bias;

    #pragma unroll
    for (int k0 = 0; k0 < HID; k0 += 4) {
        v2f a, bv;
        a[0]  = ctx[(m0 + row) * HID + k0 + khalf];
        a[1]  = ctx[(m0 + row) * HID + k0 + khalf + 1];
        bv[0] = Wo[(k0 + khalf) * VOC + n0 + col];
        bv[1] = Wo[(k0 + khalf + 1) * VOC + n0 + col];
        // (neg_a, A, neg_b, B, c_mod, C, reuse_a, reuse_b)
        c = __builtin_amdgcn_wmma_f32_16x16x4_f32(false, a, false, bv,
                                                  (short)0, c, false, false);
    }

    const int mbase = m0 + ((lane >> 4) << 3);
    const int ncol  = n0 + col;
    #pragma unroll
    for (int i = 0; i < 8; ++i) out[(mbase + i) * VOC + ncol] = c[i];
}

// ---------------------------------------------------------------------------
extern "C" void kernel_launch(void* const* d_in, const int* in_sizes, int n_in,
                              void* d_out, int out_size, void* d_ws, size_t ws_size,
                              hipStream_t stream)
{
    const int*   seq   = (const int*)  d_in[0];
    const float* embed = (const float*)d_in[1];
    const float* W1    = (const float*)d_in[2];
    const float* b1    = (const float*)d_in[3];
    const float* W2    = (const float*)d_in[4];
    const float* b2    = (const float*)d_in[5];
    const float* ln_g  = (const float*)d_in[6];
    const float* ln_b  = (const float*)d_in[7];
    const float* Ws    = (const float*)d_in[8];
    const float* bs    = (const float*)d_in[9];
    const float* We    = (const float*)d_in[10];
    const float* be    = (const float*)d_in[11];
    const float* Wo    = (const float*)d_in[12];
    const float* bo    = (const float*)d_in[13];

    float* ws  = (float*)d_ws;
    float* out = (float*)d_out;

    precompute_tables<<<dim3(VOC), dim3(128), 0, stream>>>(
        embed, W1, b1, W2, b2, ln_g, ln_b, Ws, bs, We, be, ws);

    scan_kernel<<<dim3(BB), dim3(64), 0, stream>>>(seq, ws);

    out_gemm_wmma<<<dim3(8, 4), dim3(32), 0, stream>>>(ws, Wo, bo, out);
}